// AvgPoolBackbone_83468394430600
// MI455X (gfx1250) — compile-verified
//
#include <hip/hip_runtime.h>

typedef __attribute__((ext_vector_type(2))) float v2f;
typedef __attribute__((ext_vector_type(8))) float v8f;

#define B_DIM 128
#define S_DIM 2048
#define D_DIM 256
#define SCHUNKS 8
#define CHUNK (S_DIM / SCHUNKS)   // 256 rows per block

__global__ __launch_bounds__(256) void zero_out_kernel(float* __restrict__ out, int n) {
    int i = blockIdx.x * blockDim.x + threadIdx.x;
    if (i < n) out[i] = 0.0f;
}

// One block per (batch, s-chunk). 8 waves; wave w owns d columns [32w, 32w+32).
// Masked sum via V_WMMA_F32_16X16X4_F32, accumulating over s in steps of 4.
__global__ __launch_bounds__(256) void AvgPool_wmma_kernel(
    const float* __restrict__ x,
    const long long* __restrict__ start_padding_indices,   // int64
    float* __restrict__ out) {

    const int b     = blockIdx.x;
    const int chunk = blockIdx.y;

    long long sp = start_padding_indices[b];
    const int eff = (sp < 0) ? S_DIM : (int)sp;           // effective length

    const int s_begin = chunk * CHUNK;
    if (s_begin >= eff) return;                            // whole chunk is padding (block-uniform)

    const int s_end_full = s_begin + CHUNK;
    const int s_end  = (eff < s_end_full) ? eff : s_end_full;  // rows with mask==1
    const int s_stop = (s_end + 3) & ~3;                       // round up to K=4 group (<= s_end_full)

    const float scale = 1.0f / (float)eff;                 // eff > 0 guaranteed here

    const int wave = threadIdx.x >> 5;
    const int lane = threadIdx.x & 31;
    const int hi   = lane >> 4;      // half-wave group: supplies K={0,1} or K={2,3}
    const int n    = lane & 15;      // N (d) index within tile
    const int d0   = wave * 32 + n;  // tile 0 column
    const int d1   = d0 + 16;        // tile 1 column

    const float* __restrict__ xb = x + (size_t)b * (S_DIM * D_DIM);

    v8f c0 = {};
    v8f c1 = {};

    for (int s0 = s_begin; s0 < s_stop; s0 += 4) {
        const int rA = s0 + 2 * hi;      // this lane's first K row
        const int rB = rA + 1;           // this lane's second K row

        const float* __restrict__ pA = xb + (size_t)rA * D_DIM;
        const float* __restrict__ pB = xb + (size_t)rB * D_DIM;

        // Prefetch next iteration's rows (global_prefetch_b8, speculative).
        __builtin_prefetch(pA + 4 * D_DIM, 0, 0);
        __builtin_prefetch(pB + 4 * D_DIM, 0, 0);

        // A operand: prefix mask, identical across all 16 M rows.
        v2f a;
        a.x = (rA < s_end) ? 1.0f : 0.0f;
        a.y = (rB < s_end) ? 1.0f : 0.0f;

        // B operands: 4(s) x 16(d) tiles of x. Two tiles -> one 128B line per row.
        v2f bv0, bv1;
        bv0.x = pA[d0];  bv0.y = pB[d0];
        bv1.x = pA[d1];  bv1.y = pB[d1];

        // D = A*B + C  (8 args: neg_a, A, neg_b, B, c_mod, C, reuse_a, reuse_b)
        c0 = __builtin_amdgcn_wmma_f32_16x16x4_f32(false, a, false, bv0, (short)0, c0, false, false);
        c1 = __builtin_amdgcn_wmma_f32_16x16x4_f32(false, a, false, bv1, (short)0, c1, false, false);
    }

    // Row M=0 of D lives in VGPR0 of lanes 0..15 (N = lane). All M rows are
    // identical, so the low half-wave carries everything we need.
    if (hi == 0) {
        float* o = out + (size_t)b * D_DIM;
        atomicAdd(&o[d0], c0[0] * scale);
        atomicAdd(&o[d1], c1[0] * scale);
    }
}

extern "C" void kernel_launch(void* const* d_in, const int* in_sizes, int n_in,
                              void* d_out, int out_size, void* d_ws, size_t ws_size,
                              hipStream_t stream) {
    const float*     x   = (const float*)d_in[0];
    const long long* idx = (const long long*)d_in[1];
    float*           out = (float*)d_out;

    // d_out is poisoned by the harness; zero it before accumulation.
    int nz = out_size;
    zero_out_kernel<<<(nz + 255) / 256, 256, 0, stream>>>(out, nz);

    dim3 grid(B_DIM, SCHUNKS);
    AvgPool_wmma_kernel<<<grid, 256, 0, stream>>>(x, idx, out);
}